// Net_90074054132244
// MI455X (gfx1250) — compile-verified
//
#include <hip/hip_runtime.h>

typedef __attribute__((ext_vector_type(2))) float v2f;
typedef __attribute__((ext_vector_type(8))) float v8f;

// ---------------- CSR build ----------------
__global__ void k_init(float* degf, int* cnt, int* cursor, int N) {
  int i = blockIdx.x * blockDim.x + threadIdx.x;
  if (i < N) { degf[i] = 1.0f; cnt[i] = 0; cursor[i] = 0; }  // 1.0 = self loop
}

__global__ void k_count(const int* __restrict__ dst, float* degf, int* cnt, int E) {
  int e = blockIdx.x * blockDim.x + threadIdx.x;
  if (e < E) {
    int d = dst[e];
    atomicAdd(&degf[d], 1.0f);
    atomicAdd(&cnt[d], 1);
  }
}

__global__ void k_dinv(float* dinv, int N) {
  int i = blockIdx.x * blockDim.x + threadIdx.x;
  if (i < N) {
    float d = dinv[i];
    dinv[i] = (d > 0.0f) ? rsqrtf(d) : 0.0f;
  }
}

// single-block exclusive scan (50k elements, trivial cost)
__global__ void k_scan(const int* __restrict__ cnt, int* rowptr, int N) {
  __shared__ int sh[256];
  __shared__ int base;
  if (threadIdx.x == 0) base = 0;
  __syncthreads();
  for (int start = 0; start < N; start += 256) {
    int i = start + (int)threadIdx.x;
    int v = (i < N) ? cnt[i] : 0;
    sh[threadIdx.x] = v;
    __syncthreads();
    #pragma unroll
    for (int off = 1; off < 256; off <<= 1) {
      int t = (threadIdx.x >= (unsigned)off) ? sh[threadIdx.x - off] : 0;
      __syncthreads();
      sh[threadIdx.x] += t;
      __syncthreads();
    }
    if (i < N) rowptr[i] = base + sh[threadIdx.x] - v;  // exclusive
    __syncthreads();
    if (threadIdx.x == 0) base += sh[255];
    __syncthreads();
  }
  if (threadIdx.x == 0) rowptr[N] = base;
}

__global__ void k_fill(const int* __restrict__ src, const int* __restrict__ dst,
                       const int* __restrict__ rowptr, int* cursor, int* col, int E) {
  int e = blockIdx.x * blockDim.x + threadIdx.x;
  if (e < E) {
    int d = dst[e];
    int p = rowptr[d] + atomicAdd(&cursor[d], 1);
    col[p] = src[e];
  }
}

// ---------------- fp32 WMMA GEMM: Y = X(NxK) @ W(KxM), optional bias+relu ----------------
// Block = 8 waves sharing one 64-col span of W, staged in LDS with k-pair
// interleaving: element (k,c) -> wlds[(k/2)*160 + 2c + (k&1)], so each lane's
// B operand (W[ka][col], W[ka+1][col]) is ONE aligned ds_load_b64. Pair-row
// stride 160 (== 32 mod 64 banks) makes the two half-waves (which read rows
// k/k+1 vs k+2/k+3) hit disjoint bank halves -> conflict-free.
// Each wave computes a 16x64 strip with 4 V_WMMA_F32_16X16X4_F32 per k-step.
// Inner loop is branch-free: out-of-range W columns are address-clamped and
// those lanes' stores are guarded instead (EXEC all-1s at every WMMA).
template <int KMAX>
__global__ void k_gemm_wmma(const float* __restrict__ X, const float* __restrict__ W,
                            const float* __restrict__ bias, float* __restrict__ Y,
                            int N, int K, int M, int rowTiles, int epi) {
  __shared__ float wlds[(KMAX / 2) * 160];
  const int col0 = blockIdx.y << 6;

  // cooperative stage of W[:, col0:col0+64] into LDS (clamped, branch-free)
  for (int idx = threadIdx.x; idx < (K << 6); idx += 256) {
    int kk = idx >> 6;
    int c  = idx & 63;
    int colg = col0 + c;
    colg = (colg < M) ? colg : (M - 1);
    wlds[(kk >> 1) * 160 + (c << 1) + (kk & 1)] = W[(size_t)kk * M + colg];
  }
  __syncthreads();

  int lane  = threadIdx.x & 31;
  int rt    = blockIdx.x * 8 + (threadIdx.x >> 5);
  if (rt >= rowTiles) return;                 // wave-uniform; EXEC all-1s at WMMA
  int row0  = rt << 4;
  int lrow  = lane & 15;
  int khalf = lane >> 4;

  int r = row0 + lrow;
  r = (r < N) ? r : (N - 1);                  // clamp (stores guarded)
  const float* xrow = X + (size_t)r * K;

  v8f acc0 = {}, acc1 = {}, acc2 = {}, acc3 = {};
  for (int k = 0; k < K; k += 4) {            // K multiple of 4 (500,128)
    int ka = k + (khalf << 1);
    v2f a;
    a.x = xrow[ka];
    a.y = xrow[ka + 1];
    // pair-row base: rows (ka, ka+1) live at pair index ka/2 = k/2 + khalf
    const float* lp = &wlds[((k >> 1) + khalf) * 160 + (lrow << 1)];
    v2f b0 = *(const v2f*)(lp + 0);           // one ds_load_b64 per operand
    v2f b1 = *(const v2f*)(lp + 32);
    v2f b2 = *(const v2f*)(lp + 64);
    v2f b3 = *(const v2f*)(lp + 96);
    acc0 = __builtin_amdgcn_wmma_f32_16x16x4_f32(false, a, false, b0, (short)0, acc0, false, false);
    acc1 = __builtin_amdgcn_wmma_f32_16x16x4_f32(false, a, false, b1, (short)0, acc1, false, false);
    acc2 = __builtin_amdgcn_wmma_f32_16x16x4_f32(false, a, false, b2, (short)0, acc2, false, false);
    acc3 = __builtin_amdgcn_wmma_f32_16x16x4_f32(false, a, false, b3, (short)0, acc3, false, false);
  }

  v8f accv[4] = {acc0, acc1, acc2, acc3};
  #pragma unroll
  for (int c = 0; c < 4; ++c) {
    int col = col0 + (c << 4) + lrow;
    if (col < M) {
      float bv = epi ? bias[col] : 0.0f;
      #pragma unroll
      for (int v = 0; v < 8; ++v) {
        int row = row0 + v + (khalf << 3);    // D layout: VGPR v -> M=v / v+8
        if (row < N) {
          float val = accv[c][v] + bv;
          if (epi) val = fmaxf(val, 0.0f);
          Y[(size_t)row * M + col] = val;
        }
      }
    }
  }
}

// ---------------- aggregation: one wave per node, 128 features as float4/lane ----------------
// out[i] = dinv[i]*( sum_{j->i} dinv[j]*xw[j] + dinv[i]*xw[i] ) [+ b, relu]
__global__ void k_agg128(const float* __restrict__ xw, const float* __restrict__ dinv,
                         const int* __restrict__ rowptr, const int* __restrict__ col,
                         const float* __restrict__ bias, float* __restrict__ out,
                         int N, int epi) {
  int wv   = (blockIdx.x * blockDim.x + threadIdx.x) >> 5;
  int lane = threadIdx.x & 31;
  if (wv >= N) return;
  int f = lane << 2;
  float4 acc = make_float4(0.0f, 0.0f, 0.0f, 0.0f);
  int s = rowptr[wv], e = rowptr[wv + 1];
  for (int i = s; i < e; ++i) {
    int j = col[i];
    int inext = (i + 1 < e) ? (i + 1) : i;    // branch-free prefetch of next row
    __builtin_prefetch(xw + (size_t)col[inext] * 128 + f, 0, 0);
    float wj = dinv[j];
    float4 v = *(const float4*)(xw + (size_t)j * 128 + f);
    acc.x += wj * v.x; acc.y += wj * v.y; acc.z += wj * v.z; acc.w += wj * v.w;
  }
  float di = dinv[wv];
  float4 sv = *(const float4*)(xw + (size_t)wv * 128 + f);
  float4 r;
  r.x = di * (acc.x + di * sv.x);
  r.y = di * (acc.y + di * sv.y);
  r.z = di * (acc.z + di * sv.z);
  r.w = di * (acc.w + di * sv.w);
  if (epi) {
    r.x = fmaxf(r.x + bias[f + 0], 0.0f);
    r.y = fmaxf(r.y + bias[f + 1], 0.0f);
    r.z = fmaxf(r.z + bias[f + 2], 0.0f);
    r.w = fmaxf(r.w + bias[f + 3], 0.0f);
  }
  *(float4*)(out + (size_t)wv * 128 + f) = r;
}

extern "C" void kernel_launch(void* const* d_in, const int* in_sizes, int n_in,
                              void* d_out, int out_size, void* d_ws, size_t ws_size,
                              hipStream_t stream) {
  const float* x  = (const float*)d_in[0];
  const int*   ei = (const int*)d_in[1];
  const int E = in_sizes[1] / 2;
  const int* src = ei;         // edge_index[0]
  const int* dst = ei + E;     // edge_index[1]
  const int H = in_sizes[3];   // b1 -> 128
  const int F = in_sizes[13];  // b6 -> 500
  const int N = in_sizes[0] / F;

  const float* Wt[6] = {(const float*)d_in[2],  (const float*)d_in[4],  (const float*)d_in[6],
                        (const float*)d_in[8],  (const float*)d_in[10], (const float*)d_in[12]};
  const float* bt[6] = {(const float*)d_in[3],  (const float*)d_in[5],  (const float*)d_in[7],
                        (const float*)d_in[9],  (const float*)d_in[11], (const float*)d_in[13]};

  // workspace carve-out (256B aligned): ~55 MB total
  char* ws = (char*)d_ws;
  size_t off = 0;
  auto take = [&](size_t bytes) -> char* {
    char* p = ws + off;
    off += (bytes + 255) & ~(size_t)255;
    return p;
  };
  float* dinv   = (float*)take((size_t)N * 4);
  int*   cnt    = (int*)  take((size_t)N * 4);
  int*   cursor = (int*)  take((size_t)N * 4);
  int*   rowptr = (int*)  take((size_t)(N + 1) * 4);
  int*   col    = (int*)  take((size_t)E * 4);
  float* bufA   = (float*)take((size_t)N * H * 4);
  float* bufB   = (float*)take((size_t)N * H * 4);

  int gN = (N + 255) / 256, gE = (E + 255) / 256;
  k_init <<<gN, 256, 0, stream>>>(dinv, cnt, cursor, N);
  k_count<<<gE, 256, 0, stream>>>(dst, dinv, cnt, E);
  k_dinv <<<gN, 256, 0, stream>>>(dinv, N);
  k_scan <<<1, 256, 0, stream>>>(cnt, rowptr, N);
  k_fill <<<gE, 256, 0, stream>>>(src, dst, rowptr, cursor, col, E);

  int rowTiles = (N + 15) / 16;
  dim3 blk(256, 1, 1);

  auto gemm500 = [&](const float* X, const float* W, const float* b, float* Y,
                     int K, int M, int epi) {
    dim3 grid((rowTiles + 7) / 8, (M + 63) / 64, 1);
    k_gemm_wmma<500><<<grid, blk, 0, stream>>>(X, W, b, Y, N, K, M, rowTiles, epi);
  };
  auto gemm128 = [&](const float* X, const float* W, const float* b, float* Y,
                     int K, int M, int epi) {
    dim3 grid((rowTiles + 7) / 8, (M + 63) / 64, 1);
    k_gemm_wmma<128><<<grid, blk, 0, stream>>>(X, W, b, Y, N, K, M, rowTiles, epi);
  };
  auto agg = [&](const float* XW, const float* b, float* O, int epi) {
    long threads = (long)N * 32;
    int blocks = (int)((threads + 255) / 256);
    k_agg128<<<blocks, 256, 0, stream>>>(XW, dinv, rowptr, col, b, O, N, epi);
  };

  // layer 1: GEMM 500->128 first, aggregate in 128-dim
  gemm500(x, Wt[0], bt[0], bufA, F, H, 0);
  agg(bufA, bt[0], bufB, 1);
  // layers 2..5: 128->128
  for (int l = 1; l < 5; ++l) {
    gemm128(bufB, Wt[l], bt[l], bufA, H, H, 0);
    agg(bufA, bt[l], bufB, 1);
  }
  // layer 6: aggregation is linear -> aggregate in 128-dim, then GEMM 128->500 (+b, relu)
  agg(bufB, bt[5], bufA, 0);
  gemm128(bufA, Wt[5], bt[5], (float*)d_out, H, F, 1);
}